// CentroidTDBase_60490319397195
// MI455X (gfx1250) — compile-verified
//
#include <hip/hip_runtime.h>

typedef __attribute__((ext_vector_type(2))) float v2f;
typedef __attribute__((ext_vector_type(8))) float v8f;

namespace {
constexpr int kB = 256;   // batch
constexpr int kD = 256;   // x feature dim
constexpr int kF = 512;   // latent dim
constexpr float kP = 0.05f;          // P_TARGET
constexpr float kOneMinusMom = 0.05f; // 1 - MOMENTUM
}

__device__ __forceinline__ v8f wmma4(v2f a, v2f b, v8f c) {
  // D = A(16x4,f32) x B(4x16,f32) + C(16x16,f32)
  return __builtin_amdgcn_wmma_f32_16x16x4_f32(false, a, false, b, (short)0, c,
                                               false, false);
}

// Tile (i0..i0+15, j0..j0+15) of Craw = Z^T Z, fused epilogue:
//   dmu = 0.05*(Craw/B - mu);  G = (p^2 - (mu+dmu)) * Craw
__global__ __launch_bounds__(32) void k_ztz_dmu_g(const float* __restrict__ z,
                                                  const float* __restrict__ mu,
                                                  float* __restrict__ dmu,
                                                  float* __restrict__ G) {
  const int j0 = blockIdx.x * 16;
  const int i0 = blockIdx.y * 16;
  const int lane = (int)threadIdx.x;
  const int g = lane >> 4, l = lane & 15;
  v8f acc = {};
  for (int k0 = 0; k0 < kB; k0 += 4) {
    const int ka = k0 + 2 * g;
    v2f a, b;
    a.x = z[(ka + 0) * kF + i0 + l];  // A[m=l, k=ka]   = Z[ka, i0+l]
    a.y = z[(ka + 1) * kF + i0 + l];
    b.x = z[(ka + 0) * kF + j0 + l];  // B[k=ka, n=l]   = Z[ka, j0+l]
    b.y = z[(ka + 1) * kF + j0 + l];
    acc = wmma4(a, b, acc);
  }
#pragma unroll
  for (int r = 0; r < 8; ++r) {
    const int i = i0 + r + 8 * g;
    const int j = j0 + l;
    const float craw = acc[r];
    const float m = mu[i * kF + j];
    const float d = kOneMinusMom * (craw * (1.0f / kB) - m);
    dmu[i * kF + j] = d;
    const float mu_t = m + d;
    G[i * kF + j] = (kP * kP - mu_t) * craw;
  }
}

// mu_self = diag(mu + dmu); dbias_f = dbias_p = p - mu_self
__global__ __launch_bounds__(256) void k_diag(const float* __restrict__ mu,
                                              const float* __restrict__ dmu,
                                              float* __restrict__ mu_self,
                                              float* __restrict__ dbias_f,
                                              float* __restrict__ dbias_p) {
  const int i = blockIdx.x * blockDim.x + (int)threadIdx.x;
  if (i < kF) {
    const float ms = mu[i * kF + i] + dmu[i * kF + i];
    mu_self[i] = ms;
    dbias_f[i] = kP - ms;
    dbias_p[i] = kP - ms;
  }
}

// rvec[i] = rowsum(G)[i] / (B*F);   zbar[i] = mean_b z[b,i]
// one wave32 per row i
__global__ __launch_bounds__(256) void k_rows(const float* __restrict__ G,
                                              const float* __restrict__ z,
                                              float* __restrict__ rvec,
                                              float* __restrict__ zbar) {
  const int wave = (blockIdx.x * (int)blockDim.x + (int)threadIdx.x) >> 5;
  const int lane = (int)threadIdx.x & 31;
  if (wave >= kF) return;
  float s = 0.0f;
  for (int j = lane; j < kF; j += 32) s += G[wave * kF + j];
#pragma unroll
  for (int off = 16; off > 0; off >>= 1) s += __shfl_xor(s, off, 32);
  float t = 0.0f;
  for (int b = lane; b < kB; b += 32) t += z[b * kF + wave];
#pragma unroll
  for (int off = 16; off > 0; off >>= 1) t += __shfl_xor(t, off, 32);
  if (lane == 0) {
    rvec[wave] = s * (1.0f / ((float)kB * (float)kF));
    zbar[wave] = t * (1.0f / (float)kB);
  }
}

// dkernel_p = U^T E / B
__global__ __launch_bounds__(32) void k_dwp(const float* __restrict__ u,
                                            const float* __restrict__ e,
                                            float* __restrict__ dkp) {
  const int j0 = blockIdx.x * 16;
  const int i0 = blockIdx.y * 16;
  const int lane = (int)threadIdx.x;
  const int g = lane >> 4, l = lane & 15;
  v8f acc = {};
  for (int k0 = 0; k0 < kB; k0 += 4) {
    const int ka = k0 + 2 * g;
    v2f a, b;
    a.x = u[(ka + 0) * kF + i0 + l];
    a.y = u[(ka + 1) * kF + i0 + l];
    b.x = e[(ka + 0) * kF + j0 + l];
    b.y = e[(ka + 1) * kF + j0 + l];
    acc = wmma4(a, b, acc);
  }
#pragma unroll
  for (int r = 0; r < 8; ++r) {
    const int i = i0 + r + 8 * g;
    dkp[i * kF + j0 + l] = acc[r] * (1.0f / kB);
  }
}

// dkernel_f[d,i] = (p - mu_self[i]) * (M[d,i]/B - wf[d,i]*zbar[i])
//                + wf[d,i]*rvec[i] - H[d,i]/(B*F)
// with M = X^T Z (K=B) and H = W_f * G^T (K=F), both via WMMA.
__global__ __launch_bounds__(32) void k_dwf(const float* __restrict__ x,
                                            const float* __restrict__ z,
                                            const float* __restrict__ wf,
                                            const float* __restrict__ G,
                                            const float* __restrict__ mu_self,
                                            const float* __restrict__ zbar,
                                            const float* __restrict__ rvec,
                                            float* __restrict__ dkf) {
  const int i0 = blockIdx.x * 16;  // F (columns)
  const int d0 = blockIdx.y * 16;  // D (rows)
  const int lane = (int)threadIdx.x;
  const int g = lane >> 4, l = lane & 15;

  v8f accM = {};
  for (int k0 = 0; k0 < kB; k0 += 4) {
    const int ka = k0 + 2 * g;
    v2f a, b;
    a.x = x[(ka + 0) * kD + d0 + l];  // A[m=d, k] = X[k, d]  (X^T)
    a.y = x[(ka + 1) * kD + d0 + l];
    b.x = z[(ka + 0) * kF + i0 + l];  // B[k, n=i] = Z[k, i]
    b.y = z[(ka + 1) * kF + i0 + l];
    accM = wmma4(a, b, accM);
  }

  v8f accH = {};
  for (int k0 = 0; k0 < kF; k0 += 4) {
    const int ka = k0 + 2 * g;
    v2f a, b;
    a.x = wf[(d0 + l) * kF + ka + 0];  // A[m=d, k=j] = W_f[d, j] (row-major)
    a.y = wf[(d0 + l) * kF + ka + 1];
    b.x = G[(i0 + l) * kF + ka + 0];   // B[k=j, n=i] = G[i, j]  (G^T)
    b.y = G[(i0 + l) * kF + ka + 1];
    accH = wmma4(a, b, accH);
  }

  const int i = i0 + l;  // N = l for both lane halves
  const float pm = kP - mu_self[i];
  const float zb = zbar[i];
  const float rv = rvec[i];
#pragma unroll
  for (int r = 0; r < 8; ++r) {
    const int d = d0 + r + 8 * g;
    const float w = wf[d * kF + i];
    dkf[d * kF + i] = pm * (accM[r] * (1.0f / kB) - w * zb) + w * rv
                    - accH[r] * (1.0f / ((float)kB * (float)kF));
  }
}

extern "C" void kernel_launch(void* const* d_in, const int* in_sizes, int n_in,
                              void* d_out, int out_size, void* d_ws, size_t ws_size,
                              hipStream_t stream) {
  (void)in_sizes; (void)n_in; (void)out_size; (void)ws_size;

  const float* z  = (const float*)d_in[0];
  const float* x  = (const float*)d_in[1];
  const float* u  = (const float*)d_in[2];
  const float* e  = (const float*)d_in[3];
  const float* mu = (const float*)d_in[4];
  const float* wf = (const float*)d_in[5];
  // d_in[6] = w_p_kernel: unused by the reference math.

  float* out = (float*)d_out;
  float* dmu     = out;                 // (F,F)
  float* dkf     = dmu + kF * kF;       // (D,F)
  float* dbias_f = dkf + kD * kF;       // (F,)
  float* dkp     = dbias_f + kF;        // (F,F)
  float* dbias_p = dkp + kF * kF;       // (F,)

  float* ws = (float*)d_ws;
  float* G       = ws;                  // (F,F)
  float* rvec    = G + kF * kF;         // (F,)
  float* zbar    = rvec + kF;           // (F,)
  float* mu_self = zbar + kF;           // (F,)

  k_ztz_dmu_g<<<dim3(kF / 16, kF / 16), 32, 0, stream>>>(z, mu, dmu, G);
  k_diag<<<(kF + 255) / 256, 256, 0, stream>>>(mu, dmu, mu_self, dbias_f, dbias_p);
  k_rows<<<(kF * 32 + 255) / 256, 256, 0, stream>>>(G, z, rvec, zbar);
  k_dwp<<<dim3(kF / 16, kF / 16), 32, 0, stream>>>(u, e, dkp);
  k_dwf<<<dim3(kF / 16, kD / 16), 32, 0, stream>>>(x, z, wf, G, mu_self, zbar,
                                                   rvec, dkf);
}